// Interpolator1D_78769700209151
// MI455X (gfx1250) — compile-verified
//
#include <hip/hip_runtime.h>
#include <hip/hip_bf16.h>

// Problem constants (match reference)
#define NXK 4096     // number of knots
#define NQK 262144   // number of queries
#define CK  128      // channels

typedef __attribute__((ext_vector_type(2))) float v2f;
typedef __attribute__((ext_vector_type(4))) float v4f;
typedef __attribute__((ext_vector_type(8))) float v8f;

__device__ __forceinline__ int imin(int a, int b) { return a < b ? a : b; }
__device__ __forceinline__ int imax(int a, int b) { return a > b ? a : b; }

// dxi[j] = 1/(x[j+1]-x[j]), 0 when dx==0; index clamped so callers can ask
// for out-of-range j (the corresponding band weight is selected to 0 anyway).
__device__ __forceinline__ float dxi_at(const float* __restrict__ x, int j) {
  j = imin(imax(j, 0), NXK - 2);
  float dx = x[j + 1] - x[j];
  return (dx == 0.0f) ? 0.0f : (1.0f / dx);
}

// Band-matrix weight M[r, cc] of the cubic derivative estimator:
//   fx[0]    =  dxi0*(f1-f0)
//   fx[j]    =  0.5*(dxi[j-1]*(f[j]-f[j-1]) + dxi[j]*(f[j+1]-f[j]))
//   fx[NX-1] =  dxi[NX-2]*(f[NX-1]-f[NX-2])
// Branch-free (ternary chains -> v_cndmask) so EXEC stays all-ones for WMMA.
__device__ __forceinline__ float band_weight(const float* __restrict__ x, int r, int cc) {
  float dm = dxi_at(x, r - 1);           // dxi[r-1]
  float dp = dxi_at(x, r);               // dxi[r]
  float w_int = (cc == r - 1) ? (-0.5f * dm)
              : (cc == r    ) ? (0.5f * (dm - dp))
              : (cc == r + 1) ? (0.5f * dp)
              : 0.0f;
  float w_lo  = (cc == 0) ? -dp : (cc == 1) ? dp : 0.0f;            // r == 0 (dp = dxi[0])
  float w_hi  = (cc == NXK - 2) ? -dm : (cc == NXK - 1) ? dm : 0.0f; // r == NX-1 (dm = dxi[NX-2])
  float w = (r == 0) ? w_lo : (r == NXK - 1) ? w_hi : w_int;
  return (cc >= 0 && cc < NXK) ? w : 0.0f;
}

// ---------------------------------------------------------------------------
// Kernel 1: fx = BandMatrix(x) @ f   via V_WMMA_F32_16X16X4_F32.
// One wave per 16x16 tile of fx. Row tile r0 needs f rows [r0-1, r0+16];
// we take a padded 20-column band [r0-1, r0+18] -> 5 chained WMMA k-steps.
// Out-of-band / out-of-range columns get weight 0 (loads clamped).
// ---------------------------------------------------------------------------
__global__ void interp1d_fx_wmma(const float* __restrict__ x,
                                 const float* __restrict__ f,
                                 float* __restrict__ fx) {
  const int wavesPerBlock = blockDim.x >> 5;
  const int waveId = blockIdx.x * wavesPerBlock + (threadIdx.x >> 5);
  const int lane   = threadIdx.x & 31;
  const int half   = lane >> 4;      // 0: lanes 0-15, 1: lanes 16-31
  const int lid    = lane & 15;

  const int nColTiles = CK / 16;                 // 8
  const int r0 = (waveId / nColTiles) * 16;      // 0..4080
  const int c0 = (waveId % nColTiles) * 16;      // 0..112
  const int cbase = r0 - 1;                      // first band column (may be -1)

  v8f acc = {};
  #pragma unroll
  for (int s = 0; s < 5; ++s) {
    // A fragment (16x4 f32): lane (M=lid, half): VGPR j holds K = j + 2*half.
    const int ka = 2 * half;
    v2f a;
    a.x = band_weight(x, r0 + lid, cbase + 4 * s + ka + 0);
    a.y = band_weight(x, r0 + lid, cbase + 4 * s + ka + 1);

    // B fragment (4x16 f32): lane (N=lid, half): VGPR j holds B[K=j+2*half, N].
    const int kr = cbase + 4 * s + 2 * half;
    const int rA = imin(imax(kr,     0), NXK - 1);  // clamped; weight is 0 when clamped
    const int rB = imin(imax(kr + 1, 0), NXK - 1);
    v2f b;
    b.x = f[(size_t)rA * CK + c0 + lid];
    b.y = f[(size_t)rB * CK + c0 + lid];

    acc = __builtin_amdgcn_wmma_f32_16x16x4_f32(
        /*neg_a=*/false, a, /*neg_b=*/false, b,
        /*c_mod=*/(short)0, acc, /*reuse_a=*/false, /*reuse_b=*/false);
  }

  // D layout: VGPR v, lane(lid,half) -> element (M = v + 8*half, N = lid).
  #pragma unroll
  for (int v = 0; v < 8; ++v) {
    fx[(size_t)(r0 + v + 8 * half) * CK + c0 + lid] = acc[v];
  }
}

// ---------------------------------------------------------------------------
// Kernel 2: streaming Hermite evaluation. One wave per query; each lane owns
// 4 channels (float4) -> coalesced B128 loads/stores; 512B/row fully vector.
// Write-bandwidth bound: 134 MB out @ 23.3 TB/s ~ 6 us. f/fx (4 MB) sit in L2.
// ---------------------------------------------------------------------------
__global__ void interp1d_eval(const float* __restrict__ xq,
                              const float* __restrict__ x,
                              const float* __restrict__ f,
                              const float* __restrict__ fx,
                              float* __restrict__ out) {
  const int wavesPerBlock = blockDim.x >> 5;
  const int q    = blockIdx.x * wavesPerBlock + (threadIdx.x >> 5);
  const int lane = threadIdx.x & 31;
  if (q >= NQK) return;

  const float xv = xq[q];

  // searchsorted(x, xv, side='right'): upper bound, then clip to [1, NX-1].
  int lo = 0, hi = NXK;
  while (lo < hi) {
    int mid = (lo + hi) >> 1;
    if (xv < x[mid]) hi = mid; else lo = mid + 1;
  }
  const int i = imin(imax(lo, 1), NXK - 1);

  const float* f0r = f  + (size_t)(i - 1) * CK;
  const float* f1r = f  + (size_t)i       * CK;
  const float* g0r = fx + (size_t)(i - 1) * CK;
  const float* g1r = fx + (size_t)i       * CK;
  __builtin_prefetch(f0r, 0, 0);   // global_prefetch_b8
  __builtin_prefetch(f1r, 0, 0);
  __builtin_prefetch(g0r, 0, 0);
  __builtin_prefetch(g1r, 0, 0);

  const float x0  = x[i - 1];
  const float x1  = x[i];
  const float dx  = x1 - x0;
  const float dxi = (dx == 0.0f) ? 0.0f : (1.0f / dx);
  const float t   = (xv - x0) * dxi;

  const v4f f0 = ((const v4f*)f0r)[lane];
  const v4f f1 = ((const v4f*)f1r)[lane];
  const v4f d0 = ((const v4f*)g0r)[lane] * dx;
  const v4f d1 = ((const v4f*)g1r)[lane] * dx;

  const v4f c2 = -3.0f * f0 + 3.0f * f1 - 2.0f * d0 - d1;
  const v4f c3 =  2.0f * f0 - 2.0f * f1 +        d0 + d1;
  const v4f r  = f0 + t * (d0 + t * (c2 + t * c3));

  ((v4f*)(out + (size_t)q * CK))[lane] = r;
}

extern "C" void kernel_launch(void* const* d_in, const int* in_sizes, int n_in,
                              void* d_out, int out_size, void* d_ws, size_t ws_size,
                              hipStream_t stream) {
  const float* xq = (const float*)d_in[0];   // (NQ,)
  const float* x  = (const float*)d_in[1];   // (NX,)
  const float* f  = (const float*)d_in[2];   // (NX, C)
  float* out = (float*)d_out;                // (NQ, C)
  float* fx  = (float*)d_ws;                 // (NX, C) scratch, 2 MB; fully overwritten

  // Stage 1: derivative precompute via WMMA banded matmul.
  // 256 row-tiles * 8 col-tiles = 2048 waves; 8 waves (256 thr) per block.
  interp1d_fx_wmma<<<2048 / 8, 256, 0, stream>>>(x, f, fx);

  // Stage 2: streaming evaluation, one wave per query, 8 queries per block.
  interp1d_eval<<<NQK / 8, 256, 0, stream>>>(xq, x, f, fx, out);
}